// TopKMarginLoss_49512382988773
// MI455X (gfx1250) — compile-verified
//
#include <hip/hip_runtime.h>
#include <math.h>

#define KTOP 3
#define MARGIN_F 0.2f
#define POS_WEIGHT_F 0.3f
#define EPS_F 1e-8f

typedef __attribute__((ext_vector_type(2))) float        v2f;
typedef __attribute__((ext_vector_type(8))) float        v8f;
typedef __attribute__((ext_vector_type(4))) unsigned int v4u;
typedef __attribute__((ext_vector_type(8))) int          v8i;
typedef __attribute__((ext_vector_type(4))) int          v4i;

#define ROWDW 260   // 256 data dwords + 4 pad dwords (TDM pad) -> conflict-free b64 reads
#define TILE_DW (16 * ROWDW)

__device__ __forceinline__ void top3_insert(float x, float& t0, float& t1, float& t2) {
    if (x > t0)      { t2 = t1; t1 = t0; t0 = x; }
    else if (x > t1) { t2 = t1; t1 = x; }
    else if (x > t2) { t2 = x; }
}

// TDM: one contiguous 16KB tile (16 rows x 256 f32) -> LDS, padded 4 dwords per 256.
// D# per CDNA5 ISA ch.8: group0 {count|flags, lds_addr, gaddr_lo, gaddr_hi|type=2},
// group1 {data_size=4B + pad fields, tensor_dim0=4096, tensor_dim1=1, tile_dim0=4096,
//         tile_dim1=0 (1D), dim0_stride=4096}.  2D max -> groups 2/3 zero.
// This toolchain exposes the 6-arg builtin (extra int32x8 group before cpol).
__device__ __forceinline__ void tdm_load_tile(const float* gsrc, unsigned lds_off) {
    unsigned long long ga = (unsigned long long)(uintptr_t)gsrc;
    unsigned ga_lo = __builtin_amdgcn_readfirstlane((unsigned)ga);
    unsigned ga_hi = __builtin_amdgcn_readfirstlane((unsigned)(ga >> 32));
    unsigned lo    = __builtin_amdgcn_readfirstlane(lds_off);
    v4u g0 = { 1u,                                   // count=1, user flags 0
               lo,                                   // lds_addr
               ga_lo,                                // global_addr[31:0]
               (ga_hi & 0x01FFFFFFu) | 0x80000000u };// global_addr[56:32] | type=2
    v8i g1 = { (int)0x07D20000u,   // data_size=2(4B) | pad_en | pad_interval=7(256dw) | pad_amount=3(4dw)
               (int)(4096u << 16), // tensor_dim0 = 4096  (bits 79:48)
               (int)(1u << 16),    // tensor_dim1 = 1     (bits 111:80)
               (int)(4096u << 16), // tile_dim0  = 4096   (bits 127:112)
               0,                  // tile_dim1=0, tile_dim2=0
               4096,               // tensor_dim0_stride[31:0]
               0, 0 };
    v4i z4 = {0, 0, 0, 0};
    v8i z8 = {0, 0, 0, 0, 0, 0, 0, 0};
    __builtin_amdgcn_tensor_load_to_lds(g0, g1, z4, z4, z8, 0);
}

// One workgroup (128 threads = 4 waves) per batch row b.
// Streams neg[b] (2048x256 f32, 2MB) once via TDM double-buffering;
// dots via V_WMMA_F32_16X16X4_F32 on LDS-staged tiles.
__global__ __launch_bounds__(128)
void topk_margin_phase1(const float* __restrict__ anchor,
                        const float* __restrict__ positives,
                        const float* __restrict__ negatives,
                        float* __restrict__ partial) {
    const int b    = blockIdx.x;
    const int tid  = threadIdx.x;
    const int lane = tid & 31;
    const int wave = tid >> 5;        // 0..3

    __shared__ float tileS[4 * 2 * TILE_DW];   // [wave][buf][16 rows * 260 dw] ~130KB
    __shared__ float anchorS[256];
    __shared__ float posS[4];
    __shared__ float naS;
    __shared__ float wtop[4][3];

    const float* aRow = anchor + (size_t)b * 256;
    anchorS[tid]       = aRow[tid];
    anchorS[tid + 128] = aRow[tid + 128];
    __syncthreads();

    // ---- positives: wave w handles positive p = w (P == 4) ----
    {
        const float* pRow = positives + ((size_t)b * 4 + wave) * 256;
        float dot = 0.f, pn = 0.f, an = 0.f;
        #pragma unroll
        for (int i = 0; i < 8; ++i) {
            int d = lane + 32 * i;
            float av = anchorS[d];
            float pv = pRow[d];
            dot = fmaf(av, pv, dot);
            pn  = fmaf(pv, pv, pn);
            an  = fmaf(av, av, an);
        }
        #pragma unroll
        for (int off = 16; off > 0; off >>= 1) {
            dot += __shfl_xor(dot, off);
            pn  += __shfl_xor(pn,  off);
            an  += __shfl_xor(an,  off);
        }
        float na = sqrtf(an);
        if (lane == 0) {
            posS[wave] = dot / fmaxf(na * sqrtf(pn), EPS_F);
            if (wave == 0) naS = na;
        }
    }
    __syncthreads();
    const float na = naS;

    // ---- negatives: TDM -> LDS double buffer, WMMA 16x16x4 f32 per tile ----
    const int   waveU = __builtin_amdgcn_readfirstlane(wave);
    const float* nBase = negatives + (size_t)b * 2048 * 256;
    const int row = lane & 15;   // A-matrix M row for this lane
    const int h   = lane >> 4;   // half selects K={0,1} vs K={2,3} of each chunk

    const unsigned ldsBase = (unsigned)(uintptr_t)(&tileS[0]);
    const unsigned buf0Off = ldsBase + (unsigned)(waveU * 2) * (TILE_DW * 4);
    const unsigned buf1Off = buf0Off + (TILE_DW * 4);

    // prime buffer 0 with this wave's first tile
    tdm_load_tile(nBase + (size_t)waveU * 16 * 256, buf0Off);

    float t0 = -1e30f, t1 = -1e30f, t2 = -1e30f;   // per-lane running top-3 sims
    int bufFlag = 0;

    for (int tile = waveU; tile < 128; tile += 4) {
        const float* lbase = &tileS[(waveU * 2 + bufFlag) * TILE_DW];
        if (tile + 4 < 128) {   // prefetch next tile into the other buffer
            tdm_load_tile(nBase + (size_t)(tile + 4) * 16 * 256,
                          bufFlag ? buf0Off : buf1Off);
            __builtin_amdgcn_s_wait_tensorcnt((short)1);  // current buffer ready
        } else {
            __builtin_amdgcn_s_wait_tensorcnt((short)0);
        }

        v8f acc = {0.f, 0.f, 0.f, 0.f, 0.f, 0.f, 0.f, 0.f};
        float nrm2 = 0.f;
        for (int k = 0; k < 64; ++k) {
            const int d0 = 4 * k + 2 * h;
            v2f a = *(const v2f*)(lbase + row * ROWDW + d0);  // ds_load_b64, bank-padded
            v2f bb;                                           // anchor across all 16 cols
            bb.x = anchorS[d0];
            bb.y = anchorS[d0 + 1];
            nrm2 = fmaf(a.x, a.x, nrm2);
            nrm2 = fmaf(a.y, a.y, nrm2);
            acc = __builtin_amdgcn_wmma_f32_16x16x4_f32(false, a, false, bb,
                                                        (short)0, acc, false, false);
        }
        nrm2 += __shfl_xor(nrm2, 16);   // full ||neg_row||^2 at lanes row and row+16
        #pragma unroll
        for (int r = 0; r < 8; ++r) {
            int m = r + 8 * h;                   // neg row whose dot sits in acc[r]
            float no  = sqrtf(__shfl(nrm2, m));
            float sim = acc[r] / fmaxf(na * no, EPS_F);
            top3_insert(sim, t0, t1, t2);
        }
        bufFlag ^= 1;
    }

    // merge lane 0 (rows 0-7 mod 16) with lane 16 (rows 8-15 mod 16)
    float u0 = __shfl(t0, 16), u1 = __shfl(t1, 16), u2 = __shfl(t2, 16);
    if (lane == 0) {
        top3_insert(u0, t0, t1, t2);
        top3_insert(u1, t0, t1, t2);
        top3_insert(u2, t0, t1, t2);
        wtop[wave][0] = t0; wtop[wave][1] = t1; wtop[wave][2] = t2;
    }
    __syncthreads();

    if (tid == 0) {
        float n0 = -1e30f, n1 = -1e30f, n2 = -1e30f;
        #pragma unroll
        for (int w = 0; w < 4; ++w) {
            top3_insert(wtop[w][0], n0, n1, n2);
            top3_insert(wtop[w][1], n0, n1, n2);
            top3_insert(wtop[w][2], n0, n1, n2);
        }
        const float best_neg = n0;
        float p0 = posS[0], p1 = posS[1], p2 = posS[2], p3 = posS[3];
        float best_pos = fmaxf(fmaxf(p0, p1), fmaxf(p2, p3));
        float margin_loss = fmaxf(0.f, MARGIN_F - (best_pos - best_neg));

        // stable top-3 over {pos0..pos3, n0, n1, n2}; ties favor positives
        float vals[7]  = {p0, p1, p2, p3, n0, n1, n2};
        bool  ispos[7] = {true, true, true, true, false, false, false};
        bool  used[7]  = {false, false, false, false, false, false, false};
        float topk_sum = 0.f;
        for (int kk = 0; kk < KTOP; ++kk) {
            int bi = 0; float bv = -1e38f;
            for (int j = 0; j < 7; ++j)
                if (!used[j] && vals[j] > bv) { bv = vals[j]; bi = j; }
            used[bi] = true;
            if (ispos[bi]) topk_sum += fmaxf(0.f, MARGIN_F - (vals[bi] - best_neg));
        }
        float possum = p0 + p1 + p2 + p3;
        partial[b] = margin_loss + topk_sum - (POS_WEIGHT_F * 0.25f) * possum;
    }
}

__global__ __launch_bounds__(256)
void topk_margin_phase2(const float* __restrict__ partial, float* __restrict__ out) {
    const int tid  = threadIdx.x;
    const int lane = tid & 31;
    const int wave = tid >> 5;
    __shared__ float wsum[8];
    float v = partial[tid];
    #pragma unroll
    for (int off = 16; off > 0; off >>= 1) v += __shfl_xor(v, off);
    if (lane == 0) wsum[wave] = v;
    __syncthreads();
    if (tid == 0) {
        float s = 0.f;
        #pragma unroll
        for (int w = 0; w < 8; ++w) s += wsum[w];
        out[0] = POS_WEIGHT_F + s * (1.0f / 256.0f);
    }
}

extern "C" void kernel_launch(void* const* d_in, const int* in_sizes, int n_in,
                              void* d_out, int out_size, void* d_ws, size_t ws_size,
                              hipStream_t stream) {
    (void)in_sizes; (void)n_in; (void)out_size; (void)ws_size;
    const float* anchor    = (const float*)d_in[0];   // (256, 256)
    const float* positives = (const float*)d_in[1];   // (256, 4, 256)
    const float* negatives = (const float*)d_in[2];   // (256, 2048, 256)
    float* out     = (float*)d_out;
    float* partial = (float*)d_ws;                    // 256 floats

    topk_margin_phase1<<<256, 128, 0, stream>>>(anchor, positives, negatives, partial);
    topk_margin_phase2<<<1, 256, 0, stream>>>(partial, out);
}